// GraphAttentionNetwork_45801531244889
// MI455X (gfx1250) — compile-verified
//
#include <hip/hip_runtime.h>
#include <hip/hip_bf16.h>
#include <math.h>

typedef __attribute__((ext_vector_type(2))) float v2f;
typedef __attribute__((ext_vector_type(8))) float v8f;

#define NEG_SLOPE 0.2f

__device__ __forceinline__ float leaky(float v) { return v > 0.0f ? v : NEG_SLOPE * v; }

// order-preserving float<->uint mapping for atomic max on floats
__device__ __forceinline__ unsigned f2ord(float f) {
    unsigned u = __float_as_uint(f);
    return (u & 0x80000000u) ? ~u : (u | 0x80000000u);
}
__device__ __forceinline__ float ord2f(unsigned u) {
    return (u & 0x80000000u) ? __uint_as_float(u & 0x7fffffffu) : __uint_as_float(~u);
}

// ---------------------------------------------------------------------------
// fp32 WMMA GEMM: C[M,Ncols] = op(A)[M,K] * B[K,Ncols]
// op(A) = relu(A + bias_row) when RELU_BIAS (fused inter-layer activation).
// One wave (32 lanes) computes one 16x16 tile via V_WMMA_F32_16X16X4_F32.
// ---------------------------------------------------------------------------
template<bool RELU_BIAS, int K>
__global__ void wmma_gemm_f32(const float* __restrict__ A,
                              const float* __restrict__ B,
                              const float* __restrict__ bias,
                              float* __restrict__ C,
                              int M, int Ncols)
{
    const int lane  = threadIdx.x;                       // 0..31
    const int mtile = blockIdx.x * blockDim.y + threadIdx.y;
    const int ntile = blockIdx.y;
    const int tilesM = (M + 15) >> 4;
    if (mtile >= tilesM) return;                         // wave-uniform exit

    // A frag: lanes 0-15 -> M=lane, K={k,k+1}; lanes 16-31 -> K={k+2,k+3}
    const int arow = mtile * 16 + (lane & 15);
    const int rr   = arow < M ? arow : (M - 1);          // per-lane clamp, EXEC stays full
    const int kq   = (lane >> 4) * 2;
    // B frag (dual layout): lanes 0-15 -> N=lane, K={k,k+1}; lanes 16-31 -> K={k+2,k+3}
    const int bcol = ntile * 16 + (lane & 15);

    v8f acc = {};
#pragma unroll 8
    for (int k = 0; k < K; k += 4) {
        v2f a, b;
        float a0 = A[(size_t)rr * K + k + kq];
        float a1 = A[(size_t)rr * K + k + kq + 1];
        if (RELU_BIAS) {
            a0 = fmaxf(a0 + bias[k + kq], 0.0f);
            a1 = fmaxf(a1 + bias[k + kq + 1], 0.0f);
        }
        a.x = a0; a.y = a1;
        b.x = B[(size_t)(k + kq) * Ncols + bcol];
        b.y = B[(size_t)(k + kq + 1) * Ncols + bcol];
        acc = __builtin_amdgcn_wmma_f32_16x16x4_f32(false, a, false, b,
                                                    (short)0, acc, false, false);
    }

    // C/D layout: VGPR j: lanes 0-15 -> M=j, lanes 16-31 -> M=8+j; N = lane&15
    const int ccol  = ntile * 16 + (lane & 15);
    const int rbase = mtile * 16 + ((lane >> 4) << 3);
#pragma unroll
    for (int j = 0; j < 8; ++j) {
        int row = rbase + j;
        if (row < M) C[(size_t)row * Ncols + ccol] = acc[j];
    }
}

// ---------------------------------------------------------------------------
// Per-node: attention logits alpha_src/alpha_dst, self-loop score, init max.
// ---------------------------------------------------------------------------
template<int H, int C>
__global__ void node_prep(const float* __restrict__ h,
                          const float* __restrict__ a_src,
                          const float* __restrict__ a_dst,
                          float* __restrict__ as, float* __restrict__ ad,
                          float* __restrict__ es, unsigned* __restrict__ m,
                          int nN)
{
    int t = blockIdx.x * blockDim.x + threadIdx.x;
    if (t >= nN * H) return;
    int n = t / H, hd = t % H;
    const float* hp = h + (size_t)n * H * C + hd * C;
    float s = 0.0f, d = 0.0f;
#pragma unroll 8
    for (int c = 0; c < C; ++c) {
        float v = hp[c];
        s += v * a_src[hd * C + c];
        d += v * a_dst[hd * C + c];
    }
    as[t] = s;
    ad[t] = d;
    float eself = leaky(s + d);          // self-loop edge score
    es[t] = eself;
    m[t]  = f2ord(eself);                // self loop seeds the segment max
}

// ---------------------------------------------------------------------------
// Edge pass 1: segment max of leaky(as[src]+ad[dst]) into m[dst].
// ---------------------------------------------------------------------------
template<int H>
__global__ void edge_max(const int* __restrict__ ei, int E,
                         const float* __restrict__ as, const float* __restrict__ ad,
                         unsigned* __restrict__ m)
{
    int t = blockIdx.x * blockDim.x + threadIdx.x;
    if (t >= E * H) return;
    int e = t / H, hd = t % H;
    int src = ei[e], dst = ei[E + e];
    float ee = leaky(as[src * H + hd] + ad[dst * H + hd]);
    atomicMax(&m[dst * H + hd], f2ord(ee));
}

// ---------------------------------------------------------------------------
// Per-node: decode max to float (in place) and seed denom with self-loop term.
// ---------------------------------------------------------------------------
template<int H>
__global__ void node_denom_init(unsigned* __restrict__ m_u, float* __restrict__ mf,
                                const float* __restrict__ es, float* __restrict__ dn,
                                int nN)
{
    int t = blockIdx.x * blockDim.x + threadIdx.x;
    if (t >= nN * H) return;
    float mv = ord2f(m_u[t]);
    mf[t] = mv;                           // same buffer, now holds float max
    dn[t] = __expf(es[t] - mv);           // self-loop contribution
}

// ---------------------------------------------------------------------------
// Edge pass 2: denom[dst] += exp(e - m[dst]).
// ---------------------------------------------------------------------------
template<int H>
__global__ void edge_denom(const int* __restrict__ ei, int E,
                           const float* __restrict__ as, const float* __restrict__ ad,
                           const float* __restrict__ mf, float* __restrict__ dn)
{
    int t = blockIdx.x * blockDim.x + threadIdx.x;
    if (t >= E * H) return;
    int e = t / H, hd = t % H;
    int src = ei[e], dst = ei[E + e];
    float ee = leaky(as[src * H + hd] + ad[dst * H + hd]);
    atomicAdd(&dn[dst * H + hd], __expf(ee - mf[dst * H + hd]));
}

// ---------------------------------------------------------------------------
// Per-node: initialize accumulator with the self-loop message h[i]*alpha_self.
// ---------------------------------------------------------------------------
template<int H, int C>
__global__ void node_acc_init(const float* __restrict__ h,
                              const float* __restrict__ es,
                              const float* __restrict__ mf,
                              const float* __restrict__ dn,
                              float* __restrict__ acc, int nN)
{
    size_t t = (size_t)blockIdx.x * blockDim.x + threadIdx.x;
    if (t >= (size_t)nN * H * C) return;
    int nh = (int)(t / C);                // node*H + head
    float alpha = __expf(es[nh] - mf[nh]) / (dn[nh] + 1e-16f);
    acc[t] = h[t] * alpha;
}

// ---------------------------------------------------------------------------
// Edge pass 3: acc[dst] += h[src] * alpha.  128 consecutive threads cover the
// 128 channels of one edge -> contiguous 512B gathers + contiguous L2 atomics.
// ---------------------------------------------------------------------------
template<int H, int C>
__global__ void edge_aggregate(const int* __restrict__ ei, int E,
                               const float* __restrict__ as, const float* __restrict__ ad,
                               const float* __restrict__ mf, const float* __restrict__ dn,
                               const float* __restrict__ h, float* __restrict__ acc)
{
    size_t t = (size_t)blockIdx.x * blockDim.x + threadIdx.x;
    if (t >= (size_t)E * H * C) return;
    int e   = (int)(t / (H * C));
    int rem = (int)(t % (H * C));
    int hd  = rem / C;
    int src = ei[e], dst = ei[E + e];
    float ee    = leaky(as[src * H + hd] + ad[dst * H + hd]);
    float alpha = __expf(ee - mf[dst * H + hd]) / (dn[dst * H + hd] + 1e-16f);
    float msg   = h[(size_t)src * H * C + rem] * alpha;
    atomicAdd(&acc[(size_t)dst * H * C + rem], msg);
}

// ---------------------------------------------------------------------------
// Final: out = acc2 + b2
// ---------------------------------------------------------------------------
__global__ void final_bias(const float* __restrict__ acc, const float* __restrict__ b,
                           float* __restrict__ out, int nN, int F)
{
    size_t t = (size_t)blockIdx.x * blockDim.x + threadIdx.x;
    if (t >= (size_t)nN * F) return;
    out[t] = acc[t] + b[(int)(t % F)];
}

extern "C" void kernel_launch(void* const* d_in, const int* in_sizes, int n_in,
                              void* d_out, int out_size, void* d_ws, size_t ws_size,
                              hipStream_t stream)
{
    const float* x     = (const float*)d_in[0];
    const int*   ei    = (const int*)d_in[1];     // [2, E]
    const float* W1    = (const float*)d_in[2];   // [128, 128]
    const float* asr1  = (const float*)d_in[3];   // [4, 32]
    const float* adt1  = (const float*)d_in[4];   // [4, 32]
    const float* b1    = (const float*)d_in[5];   // [128]
    const float* W2    = (const float*)d_in[6];   // [128, 64]
    const float* asr2  = (const float*)d_in[7];   // [1, 64]
    const float* adt2  = (const float*)d_in[8];   // [1, 64]
    const float* b2    = (const float*)d_in[9];   // [64]

    const int Fin = 128, F1 = 128, H1 = 4, C1 = 32, F2 = 64;
    const int nN = in_sizes[0] / Fin;             // 50000
    const int E  = in_sizes[1] / 2;               // 1600000

    // workspace layout (floats)
    float* ws = (float*)d_ws;
    size_t off = 0;
    float* h1   = ws + off; off += (size_t)nN * F1;
    float* as1  = ws + off; off += (size_t)nN * H1;
    float* ad1  = ws + off; off += (size_t)nN * H1;
    float* es1  = ws + off; off += (size_t)nN * H1;
    float* m1   = ws + off; off += (size_t)nN * H1;   // ordered-uint, then float
    float* dn1  = ws + off; off += (size_t)nN * H1;
    float* acc1 = ws + off; off += (size_t)nN * F1;
    float* h2   = ws + off; off += (size_t)nN * F2;
    float* as2  = ws + off; off += (size_t)nN;
    float* ad2  = ws + off; off += (size_t)nN;
    float* es2  = ws + off; off += (size_t)nN;
    float* m2   = ws + off; off += (size_t)nN;
    float* dn2  = ws + off; off += (size_t)nN;
    float* acc2 = ws + off; off += (size_t)nN * F2;
    (void)ws_size; (void)n_in;

    const int TB = 256;
    const dim3 gemmBlk(32, 4);                     // 4 waves, 1 tile per wave
    const int tilesM = (nN + 15) / 16;

    // ---------------- Layer 1 ----------------
    {
        dim3 grid((tilesM + 3) / 4, F1 / 16);
        wmma_gemm_f32<false, 128><<<grid, gemmBlk, 0, stream>>>(x, W1, nullptr, h1, nN, F1);
    }
    node_prep<H1, C1><<<(nN * H1 + TB - 1) / TB, TB, 0, stream>>>(
        h1, asr1, adt1, as1, ad1, es1, (unsigned*)m1, nN);
    edge_max<H1><<<(E * H1 + TB - 1) / TB, TB, 0, stream>>>(ei, E, as1, ad1, (unsigned*)m1);
    node_denom_init<H1><<<(nN * H1 + TB - 1) / TB, TB, 0, stream>>>(
        (unsigned*)m1, m1, es1, dn1, nN);
    edge_denom<H1><<<(E * H1 + TB - 1) / TB, TB, 0, stream>>>(ei, E, as1, ad1, m1, dn1);
    node_acc_init<H1, C1><<<(unsigned)(((size_t)nN * F1 + TB - 1) / TB), TB, 0, stream>>>(
        h1, es1, m1, dn1, acc1, nN);
    {
        size_t tot = (size_t)E * F1;
        edge_aggregate<H1, C1><<<(unsigned)((tot + TB - 1) / TB), TB, 0, stream>>>(
            ei, E, as1, ad1, m1, dn1, h1, acc1);
    }

    // ---------------- Layer 2 (bias+relu fused into A load) ----------------
    {
        dim3 grid((tilesM + 3) / 4, F2 / 16);
        wmma_gemm_f32<true, 128><<<grid, gemmBlk, 0, stream>>>(acc1, W2, b1, h2, nN, F2);
    }
    node_prep<1, F2><<<(nN + TB - 1) / TB, TB, 0, stream>>>(
        h2, asr2, adt2, as2, ad2, es2, (unsigned*)m2, nN);
    edge_max<1><<<(E + TB - 1) / TB, TB, 0, stream>>>(ei, E, as2, ad2, (unsigned*)m2);
    node_denom_init<1><<<(nN + TB - 1) / TB, TB, 0, stream>>>(
        (unsigned*)m2, m2, es2, dn2, nN);
    edge_denom<1><<<(E + TB - 1) / TB, TB, 0, stream>>>(ei, E, as2, ad2, m2, dn2);
    node_acc_init<1, F2><<<(unsigned)(((size_t)nN * F2 + TB - 1) / TB), TB, 0, stream>>>(
        h2, es2, m2, dn2, acc2, nN);
    {
        size_t tot = (size_t)E * F2;
        edge_aggregate<1, F2><<<(unsigned)((tot + TB - 1) / TB), TB, 0, stream>>>(
            ei, E, as2, ad2, m2, dn2, h2, acc2);
    }

    final_bias<<<(unsigned)(((size_t)nN * F2 + TB - 1) / TB), TB, 0, stream>>>(
        acc2, b2, (float*)d_out, nN, F2);
}